// Residual_Binary_24790551232744
// MI455X (gfx1250) — compile-verified
//
#include <hip/hip_runtime.h>
#include <stdint.h>

// Residual binary net: fused byte-wise XNOR/AND/OR chain with two column
// permutations (swap = rotate by W/2, transpose2 = even/odd de-interleave).
// One block per row; row staged into LDS via CDNA5 async global->LDS copy
// (ASYNCcnt path), permutations resolved in LDS, streamed NT stores out.
//
// Roofline: 256 MB read + 256 MB write at 23.3 TB/s => ~22 us floor.
// Pure data movement; no FLOPs, so the async-tensor/LDS machinery is the
// CDNA5 path that matters (WMMA has nothing to compute here).

#define W_COLS 8192
#define MID    (W_COLS / 2)
#define THREADS 256
#define ROW_BYTES (W_COLS * 4)

__device__ __forceinline__ int swapc(int j) { return (j + MID) & (W_COLS - 1); }
__device__ __forceinline__ int tcol(int j)  { return (j < MID) ? (j << 1) : (((j - MID) << 1) | 1); }

// Async copy: the 24-bit INST_OFFSET is added to BOTH the LDS address and the
// global address (CDNA5 ISA), so all passes share one LDS
// address VGPR and one global address VGPR pair.
#define ASYNC_LD(OFF)                                                         \
    asm volatile("global_load_async_to_lds_b128 %0, %1, off offset:" #OFF     \
                 :: "v"(laddr), "v"(gaddr) : "memory")

__global__ __launch_bounds__(THREADS)
void residual_binary_kernel(const int* __restrict__ x,
                            const int* __restrict__ w1,
                            const int* __restrict__ w2,
                            const int* __restrict__ b1,
                            const int* __restrict__ b2,
                            int* __restrict__ out)
{
    __shared__ int sX[W_COLS];   // 32 KB: raw row
    __shared__ int sY[W_COLS];   // 32 KB: after first xnor/and/swap/or stage

    const int tid = threadIdx.x;
    const uint64_t rowOff = (uint64_t)blockIdx.x * (uint64_t)ROW_BYTES;

    // ---- CDNA5 async bulk copy: one 32 KB row, global -> LDS ----
    // 8 issues x 256 lanes x 16 B = 32 KB; consecutive lanes touch
    // consecutive 16 B chunks (512 B contiguous per wave per issue).
    {
        const uint64_t gaddr = (uint64_t)(uintptr_t)x + rowOff + (unsigned)(tid * 16);
        const unsigned laddr = (unsigned)(uintptr_t)&sX[0] + (unsigned)(tid * 16);
        ASYNC_LD(0);
        ASYNC_LD(4096);
        ASYNC_LD(8192);
        ASYNC_LD(12288);
        ASYNC_LD(16384);
        ASYNC_LD(20480);
        ASYNC_LD(24576);
        ASYNC_LD(28672);
        // Drain this wave's ASYNCcnt before the workgroup barrier.
        asm volatile("s_wait_asynccnt 0x0" ::: "memory");
    }
    __syncthreads();

    // ---- Stage 1: Y[j] = X[j] | ( xnor(X[s(j)], w1[s(j)]) & b1[s(j)] ) ----
    // (fuses reference steps: xnor(w1), &b1, swap, |x)
#pragma unroll 4
    for (int j = tid; j < W_COLS; j += THREADS) {
        const int p = swapc(j);
        const int z = (~(sX[p] ^ w1[p]) & 0xFF) & b1[p];
        sY[j] = sX[j] | z;
    }
    __syncthreads();

    // ---- Stage 2: out[k] = Y[tcol(ok)] & ( xnor(Y[tcol(m)], w2[m]) | b2[m] )
    // with ok = tcol(k), m = s(ok).
    // (fuses: transpose2, xnor(w2), |b2, swap, &, transpose2)
    int* __restrict__ orow = out + (rowOff >> 2);
#pragma unroll 4
    for (int k = tid; k < W_COLS; k += THREADS) {
        const int ok = tcol(k);
        const int m  = swapc(ok);
        const int tv = sY[tcol(ok)];
        const int z2 = (~(sY[tcol(m)] ^ w2[m]) & 0xFF) | b2[m];
        __builtin_nontemporal_store(tv & z2, &orow[k]);
    }
}

extern "C" void kernel_launch(void* const* d_in, const int* in_sizes, int n_in,
                              void* d_out, int out_size, void* d_ws, size_t ws_size,
                              hipStream_t stream) {
    const int* x  = (const int*)d_in[0];
    const int* w1 = (const int*)d_in[1];
    const int* w2 = (const int*)d_in[2];
    const int* b1 = (const int*)d_in[3];
    const int* b2 = (const int*)d_in[4];
    int* out = (int*)d_out;

    const int rows = in_sizes[0] / W_COLS;   // 8192 for the reference shapes
    residual_binary_kernel<<<rows, THREADS, 0, stream>>>(x, w1, w2, b1, b2, out);
}